// YOLOF_69552700391790
// MI455X (gfx1250) — compile-verified
//
#include <hip/hip_runtime.h>
#include <hip/hip_bf16.h>
#include <stdint.h>

typedef float v2f __attribute__((ext_vector_type(2)));
typedef float v8f __attribute__((ext_vector_type(8)));

#define NUM_CLASSES 80
#define TOPK        1000
#define PAD_TOPK    1024
#define CONF_THRESH 0.05f
#define NMS_THRESH  0.6f

#define NBINS   4096
#define BIN_LO  (-3.0f)
#define BIN_INV 256.0f        // NBINS / 16.0 range
#define BIN_W   (1.0f / BIN_INV)
#define CAND_CAP 4096

// ---- workspace layout (bytes) ----
#define OFF_HIST   0            // uint[4096]            (16384 B)
#define OFF_CNT    16384        // uint  candidate count
#define OFF_THR    16388        // float logit threshold
#define OFF_MAXC   16396        // float maxcoord+1
#define OFF_CAND   16416        // u64[4096]             (32768 B)
#define OFF_BBOX   49184        // float4[1024] raw boxes
#define OFF_OBOX   65568        // float4[1024] class-offset boxes
#define OFF_AREA   81952        // float[1024]
#define OFF_SCORE  86048        // float[1024]
#define OFF_LABEL  90144        // int[1024]
#define OFF_SUPP   94240        // uint[32] initial suppressed (= !valid)
#define OFF_OVER   94464        // uint[1024*32] overlap bitmatrix (131072 B)

// zero word ranges: [0, 4104) and [23560, 56384)
#define ZW0_START 0
#define ZW0_N     4104
#define ZW1_START 23560
#define ZW1_N     32824

__global__ void yolof_zero_kernel(unsigned int* base, int startWord, int nWords) {
    int stride = gridDim.x * blockDim.x;
    for (int i = blockIdx.x * blockDim.x + threadIdx.x; i < nWords; i += stride)
        base[startWord + i] = 0u;
}

// Pass 1: histogram of logits >= BIN_LO (sigmoid is monotonic -> select on logits)
__global__ void yolof_hist_kernel(const float4* __restrict__ cls4, int nV4,
                                  unsigned int* __restrict__ hist) {
    __shared__ unsigned int h[NBINS];
    for (int i = threadIdx.x; i < NBINS; i += blockDim.x) h[i] = 0u;
    __syncthreads();
    int stride = gridDim.x * blockDim.x;
    for (int i = blockIdx.x * blockDim.x + threadIdx.x; i < nV4; i += stride) {
        int pf = i + 4 * stride;
        if (pf < nV4) __builtin_prefetch(&cls4[pf], 0, 0);   // global_prefetch_b8
        float4 v = cls4[i];
        float xs[4] = {v.x, v.y, v.z, v.w};
#pragma unroll
        for (int c = 0; c < 4; ++c) {
            float x = xs[c];
            if (x >= BIN_LO) {
                int b = (int)((x - BIN_LO) * BIN_INV);
                b = b > (NBINS - 1) ? (NBINS - 1) : b;
                atomicAdd(&h[b], 1u);
            }
        }
    }
    __syncthreads();
    for (int i = threadIdx.x; i < NBINS; i += blockDim.x)
        if (h[i]) atomicAdd(&hist[i], h[i]);
}

// Pass 2: suffix-scan histogram from the top; pick threshold bin admitting >= TOPK
__global__ void yolof_scan_kernel(const unsigned int* __restrict__ hist,
                                  float* __restrict__ thrP) {
    if (threadIdx.x == 0 && blockIdx.x == 0) {
        unsigned int cum = 0;
        int thrBin = 0;
        for (int b = NBINS - 1; b >= 0; --b) {
            cum += hist[b];
            if (cum >= TOPK) { thrBin = b; break; }
        }
        *thrP = BIN_LO + (float)thrBin * BIN_W;
    }
}

// Pass 3: collect (sortable_logit_hi32 | ~idx_lo32) keys above threshold
__global__ void yolof_collect_kernel(const float4* __restrict__ cls4, int nV4,
                                     const float* __restrict__ thrP,
                                     unsigned long long* __restrict__ cand,
                                     unsigned int* __restrict__ cnt) {
    float thr = *thrP;
    int stride = gridDim.x * blockDim.x;
    for (int i = blockIdx.x * blockDim.x + threadIdx.x; i < nV4; i += stride) {
        int pf = i + 4 * stride;
        if (pf < nV4) __builtin_prefetch(&cls4[pf], 0, 0);
        float4 v = cls4[i];
        float xs[4] = {v.x, v.y, v.z, v.w};
#pragma unroll
        for (int c = 0; c < 4; ++c) {
            float x = xs[c];
            if (x >= thr) {
                unsigned int pos = atomicAdd(cnt, 1u);
                if (pos < CAND_CAP) {
                    unsigned int fb = __float_as_uint(x);
                    fb = (fb & 0x80000000u) ? ~fb : (fb | 0x80000000u);  // order-preserving
                    unsigned int idx = (unsigned int)(i * 4 + c);
                    cand[pos] = ((unsigned long long)fb << 32) |
                                (unsigned long long)(~idx);              // ties: lowest idx first
                }
            }
        }
    }
}

// Pass 4: single-block bitonic sort (descending) + gather boxes + per-class offset prep
__global__ __launch_bounds__(1024)
void yolof_sort_topk_kernel(const unsigned long long* __restrict__ cand,
                            const unsigned int* __restrict__ cntP,
                            const float4* __restrict__ boxPred,
                            float4* __restrict__ bboxraw, float4* __restrict__ obbox,
                            float* __restrict__ areaOut, float* __restrict__ scoreOut,
                            int* __restrict__ labelOut, unsigned int* __restrict__ suppInit,
                            float* __restrict__ maxcWs) {
    __shared__ unsigned long long keys[CAND_CAP];
    __shared__ float red[1024];
    int t = threadIdx.x;
    int n = (int)*cntP;
    n = n > CAND_CAP ? CAND_CAP : n;
    for (int i = t; i < CAND_CAP; i += 1024) keys[i] = (i < n) ? cand[i] : 0ull;
    __syncthreads();
    for (int k = 2; k <= CAND_CAP; k <<= 1) {
        for (int j = k >> 1; j > 0; j >>= 1) {
            for (int i = t; i < CAND_CAP; i += 1024) {
                int ij = i ^ j;
                if (ij > i) {
                    unsigned long long a = keys[i], b = keys[ij];
                    bool descSeg = ((i & k) == 0);
                    if ((a < b) == descSeg) { keys[i] = b; keys[ij] = a; }
                }
            }
            __syncthreads();
        }
    }
    unsigned long long key = keys[t];
    unsigned int hi  = (unsigned int)(key >> 32);
    unsigned int idx = ~((unsigned int)key);
    unsigned int fb  = (hi & 0x80000000u) ? (hi & 0x7FFFFFFFu) : ~hi;
    float logit = __uint_as_float(fb);
    bool has = (t < n) && (t < TOPK);
    if (!has) { logit = -1e30f; idx = 0u; }
    float sc    = 1.0f / (1.0f + __expf(-logit));
    int   label = (int)(idx % NUM_CLASSES);
    int   anch  = (int)(idx / NUM_CLASSES);
    float4 bx = boxPred[anch];
    // maxcoord over gathered top-k boxes
    red[t] = (t < TOPK) ? fmaxf(fmaxf(bx.x, bx.y), fmaxf(bx.z, bx.w)) : -1e30f;
    __syncthreads();
    for (int s = 512; s > 0; s >>= 1) {
        if (t < s) red[t] = fmaxf(red[t], red[t + s]);
        __syncthreads();
    }
    float offC = red[0] + 1.0f;
    float lf = (float)label * offC;
    bboxraw[t]  = bx;
    obbox[t]    = make_float4(bx.x + lf, bx.y + lf, bx.z + lf, bx.w + lf);
    areaOut[t]  = (bx.z - bx.x) * (bx.w - bx.y);
    scoreOut[t] = sc;
    labelOut[t] = label;
    bool validb = has && (sc > CONF_THRESH);
    unsigned int bal = __builtin_amdgcn_ballot_w32(!validb);
    if ((t & 31) == 0) suppInit[t >> 5] = bal;
    if (t == 0) *maxcWs = offC;
}

// Pass 5: pairwise IoU over 1024x1024 (16x16 tiles, one wave each).
// union-base matrix S[m][n] = area_row[m] + area_col[n] via V_WMMA_F32_16X16X4_F32:
//   A(16x4) = [area_r | 1 | 0 | 0], B(4x16) = [1 ; area_c ; 0 ; 0]
// Bodies manually unrolled with CONSTANT d-vector indices so exactly one WMMA
// executes per tile (runtime-indexed d[v] previously forced a dynamic loop that
// re-issued the WMMA 8x). Division replaced by multiply-compare:
//   inter/max(uni,eps) > T  <=>  inter > T*max(uni,eps)   (denominator > 0)
__global__ __launch_bounds__(256)
void yolof_iou_kernel(const float4* __restrict__ obbox, const float* __restrict__ area,
                      unsigned int* __restrict__ over) {
    int wave = threadIdx.x >> 5;
    int lane = threadIdx.x & 31;
    int tile = blockIdx.x * 8 + wave;          // 4096 tiles total
    int tr = tile >> 6, tc = tile & 63;
    int r0 = tr * 16, c0 = tc * 16;
    __shared__ float4 rbox[8][16], cbox[8][16];
    __shared__ float  rarea[8][16], carea[8][16];
    if (lane < 16) {
        rbox[wave][lane] = obbox[r0 + lane];
        rarea[wave][lane] = area[r0 + lane];
    } else {
        int l = lane - 16;
        cbox[wave][l] = obbox[c0 + l];
        carea[wave][l] = area[c0 + l];
    }
    __syncthreads();
    // f32 16x4 A layout: lanes 0-15 hold K=0 (vgpr0) / K=1 (vgpr1); lanes 16-31 hold K=2/3
    v2f a, b;
    if (lane < 16) {
        a.x = rarea[wave][lane]; a.y = 1.0f;                 // A[m][0]=area_r, A[m][1]=1
        b.x = 1.0f;              b.y = carea[wave][lane];    // B[0][n]=1, B[1][n]=area_c
    } else {
        a.x = 0.0f; a.y = 0.0f; b.x = 0.0f; b.y = 0.0f;      // K=2,3 zero-pad
    }
    v8f cacc = {};
    v8f d = __builtin_amdgcn_wmma_f32_16x16x4_f32(false, a, false, b,
                                                  (short)0, cacc, false, false);
    int nn = lane & 15;
    int mh = (lane >> 4) * 8;
    float4 cb = cbox[wave][nn];
    int word = tc >> 1;
    int sh = (tc & 1) * 16;

#define IOU_BODY(V)                                                                 \
    {                                                                               \
        int m = (V) + mh;                                                           \
        float4 rb = rbox[wave][m];                                                  \
        float ltx = fmaxf(rb.x, cb.x), lty = fmaxf(rb.y, cb.y);                     \
        float rbx = fminf(rb.z, cb.z), rby = fminf(rb.w, cb.w);                     \
        float w = fmaxf(rbx - ltx, 0.0f), h = fmaxf(rby - lty, 0.0f);               \
        float inter = w * h;                                                        \
        float uni = d[(V)] - inter;                                                 \
        bool ov = (inter > NMS_THRESH * fmaxf(uni, 1e-9f)) &&                       \
                  ((r0 + m) != (c0 + nn));                                          \
        unsigned int bal = __builtin_amdgcn_ballot_w32(ov);                         \
        if (lane == 0) {                                                            \
            atomicOr(&over[(r0 + (V)) * 32 + word], (bal & 0xFFFFu) << sh);         \
            atomicOr(&over[(r0 + (V) + 8) * 32 + word],                             \
                     ((bal >> 16) & 0xFFFFu) << sh);                                \
        }                                                                           \
    }

    IOU_BODY(0) IOU_BODY(1) IOU_BODY(2) IOU_BODY(3)
    IOU_BODY(4) IOU_BODY(5) IOU_BODY(6) IOU_BODY(7)
#undef IOU_BODY
}

// Pass 6: greedy sequential NMS + final output assembly
__global__ __launch_bounds__(1024)
void yolof_nms_out_kernel(const unsigned int* __restrict__ over,
                          const unsigned int* __restrict__ suppInit,
                          const float4* __restrict__ bboxraw,
                          const float* __restrict__ score,
                          const int* __restrict__ label,
                          float* __restrict__ out) {
    __shared__ unsigned int supp[32];
    int t = threadIdx.x;
    if (t < 32) supp[t] = suppInit[t];
    __syncthreads();
    for (int i = 0; i < TOPK; ++i) {
        bool keep_i = ((supp[i >> 5] >> (i & 31)) & 1u) == 0u;  // bit i never set this iter
        if (keep_i && t < 32) {
            unsigned int ov = over[i * 32 + t];
            int wbase = t * 32;
            unsigned int m;
            if (wbase > i) m = 0xFFFFFFFFu;
            else {
                int s = i - wbase + 1;                           // suppress only j > i
                m = (s >= 32) ? 0u : (0xFFFFFFFFu << s);
            }
            unsigned int add = ov & m;
            if (add) supp[t] |= add;
        }
        __syncthreads();
    }
    if (t < TOPK) {
        bool keep = ((supp[t >> 5] >> (t & 31)) & 1u) == 0u;
        float k = keep ? 1.0f : 0.0f;
        float4 bx = bboxraw[t];
        out[t * 4 + 0] = bx.x * k;
        out[t * 4 + 1] = bx.y * k;
        out[t * 4 + 2] = bx.z * k;
        out[t * 4 + 3] = bx.w * k;
        out[4 * TOPK + t] = score[t] * k;
        out[5 * TOPK + t] = (float)label[t];
        out[6 * TOPK + t] = k;
    }
}

extern "C" void kernel_launch(void* const* d_in, const int* in_sizes, int n_in,
                              void* d_out, int out_size, void* d_ws, size_t ws_size,
                              hipStream_t stream) {
    const float* cls = (const float*)d_in[0];     // [1, N, 80] fp32
    const float* box = (const float*)d_in[1];     // [1, N, 4]  fp32
    int nCls = in_sizes[0];                       // 65,536,000
    int nV4 = nCls / 4;

    uint8_t* ws = (uint8_t*)d_ws;
    unsigned int* ws32      = (unsigned int*)ws;
    unsigned int* hist      = (unsigned int*)(ws + OFF_HIST);
    unsigned int* cnt       = (unsigned int*)(ws + OFF_CNT);
    float*        thr       = (float*)(ws + OFF_THR);
    float*        maxc      = (float*)(ws + OFF_MAXC);
    unsigned long long* cand = (unsigned long long*)(ws + OFF_CAND);
    float4* bboxraw         = (float4*)(ws + OFF_BBOX);
    float4* obbox           = (float4*)(ws + OFF_OBOX);
    float*  area            = (float*)(ws + OFF_AREA);
    float*  score           = (float*)(ws + OFF_SCORE);
    int*    label           = (int*)(ws + OFF_LABEL);
    unsigned int* suppInit  = (unsigned int*)(ws + OFF_SUPP);
    unsigned int* over      = (unsigned int*)(ws + OFF_OVER);

    // re-zero counters/histogram/over-matrix every call (ws is poisoned once, never restored)
    yolof_zero_kernel<<<32, 256, 0, stream>>>(ws32, ZW0_START, ZW0_N);
    yolof_zero_kernel<<<64, 256, 0, stream>>>(ws32, ZW1_START, ZW1_N);

    yolof_hist_kernel<<<2048, 256, 0, stream>>>((const float4*)cls, nV4, hist);
    yolof_scan_kernel<<<1, 32, 0, stream>>>(hist, thr);
    yolof_collect_kernel<<<2048, 256, 0, stream>>>((const float4*)cls, nV4, thr, cand, cnt);
    yolof_sort_topk_kernel<<<1, 1024, 0, stream>>>(cand, cnt, (const float4*)box,
                                                   bboxraw, obbox, area, score, label,
                                                   suppInit, maxc);
    yolof_iou_kernel<<<512, 256, 0, stream>>>(obbox, area, over);
    yolof_nms_out_kernel<<<1, 1024, 0, stream>>>(over, suppInit, bboxraw, score, label,
                                                 (float*)d_out);
}